// ResidueEmbedding_5437428597142
// MI455X (gfx1250) — compile-verified
//
#include <hip/hip_runtime.h>

#define N_AA   21
#define NATOM  15
#define DFEAT  128
#define DCOORD (N_AA * NATOM * 3)   // 945
#define DDIH   39
#define DH1    256
#define H1S    260                  // padded LDS stride (bank-conflict free: 260 % 64 = 4)
#define H2S    132                  // padded LDS stride (132 % 64 = 4)

typedef __attribute__((ext_vector_type(2))) float v2f;
typedef __attribute__((ext_vector_type(8))) float v8f;

// ---------------------------------------------------------------------------
// Precompute aa_proj[21][256] = aa_emb @ W1[0:128,:] and
//            chain_proj[10][256] = chain_emb @ W1[1112:1240,:]
// into workspace. 31 blocks x 256 threads; trivial cost (~1 MFLOP total).
// ---------------------------------------------------------------------------
__global__ void precompute_proj(const float* __restrict__ aa_emb,
                                const float* __restrict__ chain_emb,
                                const float* __restrict__ W1,
                                float* __restrict__ proj) {
  const int r = blockIdx.x;      // 0..30: 0..20 = aa rows, 21..30 = chain rows
  const int c = threadIdx.x;     // 0..255
  const float* emb;
  const float* wrow;
  if (r < N_AA) {
    emb  = aa_emb + (size_t)r * DFEAT;
    wrow = W1;                                            // rows [0,128)
  } else {
    emb  = chain_emb + (size_t)(r - N_AA) * DFEAT;
    wrow = W1 + (size_t)(DFEAT + DCOORD + DDIH) * DH1;    // rows [1112,1240)
  }
  float acc = 0.f;
  for (int k = 0; k < DFEAT; ++k)
    acc = fmaf(emb[k], wrow[(size_t)k * DH1 + c], acc);
  proj[(size_t)r * DH1 + c] = acc;
}

// ---------------------------------------------------------------------------
// Fused: feature build + layer1 (sparse/gathered, VALU) + layers 2-4 (WMMA f32)
// Block = 128 threads (4 wave32), tile = 16 tokens.
// ---------------------------------------------------------------------------
__global__ __launch_bounds__(128) void residue_embed_fused(
    const int*   __restrict__ seq_idx,
    const float* __restrict__ xyz,        // [NT,15,3]
    const float* __restrict__ orient,     // [NT,3,3]
    const float* __restrict__ dihedrals,  // [NT,3]
    const int*   __restrict__ chain_idx,
    const float* __restrict__ atom_mask,  // [NT,15]
    const float* __restrict__ W1, const float* __restrict__ b1,
    const float* __restrict__ W2, const float* __restrict__ b2,
    const float* __restrict__ W3, const float* __restrict__ b3,
    const float* __restrict__ W4, const float* __restrict__ b4,
    const float* __restrict__ proj,       // [31][256] aa_proj then chain_proj
    float*       __restrict__ out)        // [NT,128]
{
  __shared__ float s_h1[16][H1S];
  __shared__ float s_h2[16][H2S];
  __shared__ float s_h3[16][H2S];
  __shared__ float s_xyzl[16][48];   // 45 used
  __shared__ float s_dih[16][40];    // 39 used
  __shared__ int   s_seq[16];
  __shared__ int   s_ch[16];

  const int tid  = threadIdx.x;
  const int tok0 = blockIdx.x * 16;

  if (tid < 16) {
    s_seq[tid] = seq_idx[tok0 + tid];
    s_ch[tid]  = chain_idx[tok0 + tid];
  }

  // --- dihedral angular encoding: per angle [x, sin(f*x)x6, cos(f*x)x6] ---
  if (tid < 48) {
    const int   t = tid / 3, d = tid % 3;
    const float x = dihedrals[(size_t)(tok0 + t) * 3 + d];
    const float kF[6] = {1.f, 2.f, 3.f, 1.f, 0.5f, 1.f / 3.f};
    const int base = d * 13;
    s_dih[t][base] = x;
#pragma unroll
    for (int f = 0; f < 6; ++f) {
      s_dih[t][base + 1 + f] = sinf(kF[f] * x);
      s_dih[t][base + 7 + f] = cosf(kF[f] * x);
    }
  }

  // --- xyz_local = (R^T (xyz - xyz_CA)) * mask : 240 tasks ---
  for (int task = tid; task < 16 * NATOM; task += 128) {
    const int t = task / NATOM, a = task % NATOM;
    const int tok = tok0 + t;
    const float* p  = xyz + ((size_t)tok * NATOM + a) * 3;
    const float* ca = xyz + ((size_t)tok * NATOM + 1) * 3;   // CA = atom 1
    const float rx = p[0] - ca[0], ry = p[1] - ca[1], rz = p[2] - ca[2];
    const float* R = orient + (size_t)tok * 9;               // R[j][i] row-major
    const float m = atom_mask[(size_t)tok * NATOM + a];
    // local_i = sum_j R[j][i] * r_j  (R^T applied)
    s_xyzl[t][a * 3 + 0] = (R[0] * rx + R[3] * ry + R[6] * rz) * m;
    s_xyzl[t][a * 3 + 1] = (R[1] * rx + R[4] * ry + R[7] * rz) * m;
    s_xyzl[t][a * 3 + 2] = (R[2] * rx + R[5] * ry + R[8] * rz) * m;
  }
  __syncthreads();

  // --- layer 1: gathered sparse GEMV (45 coord rows by seq type + 39 dih rows)
  //     plus precomputed aa/chain projections. L2-resident weight gathers. ---
  const float* w1dih = W1 + (size_t)(DFEAT + DCOORD) * DH1;
  for (int t = 0; t < 16; ++t) {
    const int sq = s_seq[t], ch = s_ch[t];
    const float* w1c = W1 + (size_t)(DFEAT + sq * 45) * DH1;
    const float* pa  = proj + (size_t)sq * DH1;
    const float* pc  = proj + (size_t)(N_AA + ch) * DH1;
#pragma unroll
    for (int pass = 0; pass < 2; ++pass) {
      const int c = pass * 128 + tid;
      float acc = b1[c] + pa[c] + pc[c];
      for (int k = 0; k < 45; ++k)
        acc = fmaf(s_xyzl[t][k], w1c[(size_t)k * DH1 + c], acc);
      for (int k = 0; k < DDIH; ++k)
        acc = fmaf(s_dih[t][k], w1dih[(size_t)k * DH1 + c], acc);
      s_h1[t][c] = fmaxf(acc, 0.f);
    }
  }
  __syncthreads();

  // --- WMMA setup (wave32): lane roles per ISA 16x16 f32 layouts ---
  const int wave = tid >> 5;
  const int lane = tid & 31;
  const int half = lane >> 4;     // 0: lanes 0-15, 1: lanes 16-31
  const int lm   = lane & 15;

  // --- layer 2: h2 = relu(h1[16x256] @ W2[256x128] + b2), K=256 ---
#pragma unroll
  for (int tt = 0; tt < 2; ++tt) {
    const int col0 = (wave * 2 + tt) * 16;
    v8f acc = {};
    for (int k = 0; k < 256; k += 4) {
      v2f a = *(const v2f*)&s_h1[lm][k + 2 * half];          // A[m=lm][k+2h, +1]
      v2f b;
      b.x = W2[(size_t)(k + 2 * half)     * DFEAT + col0 + lm]; // B[k+2h][n]
      b.y = W2[(size_t)(k + 2 * half + 1) * DFEAT + col0 + lm];
      acc = __builtin_amdgcn_wmma_f32_16x16x4_f32(
          false, a, false, b, (short)0, acc, false, false);
    }
#pragma unroll
    for (int r = 0; r < 8; ++r) {
      const float v = acc[r] + b2[col0 + lm];                // D[r+8h][n]
      s_h2[r + 8 * half][col0 + lm] = fmaxf(v, 0.f);
    }
  }
  __syncthreads();

  // --- layer 3: h3 = relu(h2[16x128] @ W3[128x128] + b3), K=128 ---
#pragma unroll
  for (int tt = 0; tt < 2; ++tt) {
    const int col0 = (wave * 2 + tt) * 16;
    v8f acc = {};
    for (int k = 0; k < 128; k += 4) {
      v2f a = *(const v2f*)&s_h2[lm][k + 2 * half];
      v2f b;
      b.x = W3[(size_t)(k + 2 * half)     * DFEAT + col0 + lm];
      b.y = W3[(size_t)(k + 2 * half + 1) * DFEAT + col0 + lm];
      acc = __builtin_amdgcn_wmma_f32_16x16x4_f32(
          false, a, false, b, (short)0, acc, false, false);
    }
#pragma unroll
    for (int r = 0; r < 8; ++r) {
      const float v = acc[r] + b3[col0 + lm];
      s_h3[r + 8 * half][col0 + lm] = fmaxf(v, 0.f);
    }
  }
  __syncthreads();

  // --- layer 4: out = h3[16x128] @ W4[128x128] + b4 (no relu) ---
#pragma unroll
  for (int tt = 0; tt < 2; ++tt) {
    const int col0 = (wave * 2 + tt) * 16;
    v8f acc = {};
    for (int k = 0; k < 128; k += 4) {
      v2f a = *(const v2f*)&s_h3[lm][k + 2 * half];
      v2f b;
      b.x = W4[(size_t)(k + 2 * half)     * DFEAT + col0 + lm];
      b.y = W4[(size_t)(k + 2 * half + 1) * DFEAT + col0 + lm];
      acc = __builtin_amdgcn_wmma_f32_16x16x4_f32(
          false, a, false, b, (short)0, acc, false, false);
    }
#pragma unroll
    for (int r = 0; r < 8; ++r) {
      out[(size_t)(tok0 + r + 8 * half) * DFEAT + col0 + lm] = acc[r] + b4[col0 + lm];
    }
  }
}

// ---------------------------------------------------------------------------
extern "C" void kernel_launch(void* const* d_in, const int* in_sizes, int n_in,
                              void* d_out, int out_size, void* d_ws, size_t ws_size,
                              hipStream_t stream) {
  const int*   seq    = (const int*)  d_in[0];
  const float* xyz    = (const float*)d_in[1];
  const float* orient = (const float*)d_in[2];
  const float* dih    = (const float*)d_in[3];
  const int*   chain  = (const int*)  d_in[4];
  const float* amask  = (const float*)d_in[5];
  const float* aa_emb = (const float*)d_in[6];
  const float* ch_emb = (const float*)d_in[7];
  const float* W1 = (const float*)d_in[8];
  const float* b1 = (const float*)d_in[9];
  const float* W2 = (const float*)d_in[10];
  const float* b2 = (const float*)d_in[11];
  const float* W3 = (const float*)d_in[12];
  const float* b3 = (const float*)d_in[13];
  const float* W4 = (const float*)d_in[14];
  const float* b4 = (const float*)d_in[15];
  float* out  = (float*)d_out;
  float* proj = (float*)d_ws;          // 31*256 floats = 31.75 KB of workspace

  const int ntok = in_sizes[0];        // B*L = 32768

  precompute_proj<<<N_AA + 10, DH1, 0, stream>>>(aa_emb, ch_emb, W1, proj);
  residue_embed_fused<<<ntok / 16, 128, 0, stream>>>(
      seq, xyz, orient, dih, chain, amask,
      W1, b1, W2, b2, W3, b3, W4, b4, proj, out);
}